// LemMoEV3_21062519620177
// MI455X (gfx1250) — compile-verified
//
#include <hip/hip_runtime.h>
#include <hip/hip_bf16.h>
#include <math.h>

#define N_NODES   20000
#define N_EDGES   500000
#define NBASIS    8
#define ONEHOT    128
#define DIN       136         // ONEHOT + NBASIS
#define LATENT    128
#define WENV_N    96          // 3*MUL
#define EFDIM     288         // MUL*(1+3+5)
#define NTILES    (N_EDGES / 16)
#define WAVES_PB  4
#define TPB       (WAVES_PB * 32)
#define NBLOCKS   512

typedef float v2f __attribute__((ext_vector_type(2)));
typedef float v4f __attribute__((ext_vector_type(4)));
typedef float v8f __attribute__((ext_vector_type(8)));
typedef unsigned int u32x4 __attribute__((ext_vector_type(4)));
typedef int i32x4 __attribute__((ext_vector_type(4)));
typedef int i32x8 __attribute__((ext_vector_type(8)));

#define HAVE_TDM __has_builtin(__builtin_amdgcn_tensor_load_to_lds)

// hardware v_rcp_f32 (~1 ulp) -- avoids IEEE div expansion in silu
__device__ __forceinline__ float fast_rcp(float x) {
  return __builtin_amdgcn_rcpf(x);
}

// ---------------------------------------------------------------------------
// Tensor Data Mover: 2D tile load global -> LDS (descriptor per ISA 08_async).
//  group0: [1:0]=count=1, [63:32]=lds_addr, [120:64]=global_addr, [127:126]=2
//  group1: dw0 = flags (data_size=4B at [17:16], pad_enable [20],
//                pad_interval [24:22], pad_amount [31:25])
//          tensor_dim0 [79:48], tensor_dim1 [111:80], tile_dim0 [127:112],
//          tile_dim1 [143:128], tensor_dim0_stride [207:160]
// ---------------------------------------------------------------------------
#if HAVE_TDM
__device__ __forceinline__ void tdm_load_2d(unsigned lds_addr, const void* gptr,
                                            unsigned flags,
                                            unsigned tdim0, unsigned tdim1,
                                            unsigned tile0, unsigned tile1,
                                            unsigned stride0) {
  unsigned long long ga = (unsigned long long)(uintptr_t)gptr;
  u32x4 g0;
  g0.x = 1u;                                               // count=1
  g0.y = lds_addr;                                         // LDS byte offset
  g0.z = (unsigned)(ga & 0xffffffffull);                   // global_addr lo
  g0.w = (unsigned)((ga >> 32) & 0x01ffffffull) | (2u << 30); // hi + type=2
  i32x8 g1;
  g1[0] = (int)flags;
  g1[1] = (int)((tdim0 & 0xffffu) << 16);
  g1[2] = (int)(((tdim0 >> 16) & 0xffffu) | ((tdim1 & 0xffffu) << 16));
  g1[3] = (int)(((tdim1 >> 16) & 0xffffu) | ((tile0 & 0xffffu) << 16));
  g1[4] = (int)(tile1 & 0xffffu);                          // tile_dim2 = 0
  g1[5] = (int)stride0;                                    // stride lo32
  g1[6] = 0;
  g1[7] = 0;
  i32x4 z4 = {0, 0, 0, 0};
#if __clang_major__ >= 23
  i32x8 z8 = {0, 0, 0, 0, 0, 0, 0, 0};
  __builtin_amdgcn_tensor_load_to_lds(g0, g1, z4, z4, z8, 0);
#else
  __builtin_amdgcn_tensor_load_to_lds(g0, g1, z4, z4, 0);
#endif
}
#endif

// ---------------------------------------------------------------------------
// Row of NT 16x16 output tiles via V_WMMA_F32_16X16X4_F32, K-outer so each
// A fragment is loaded from LDS once and reused by all NT accumulators.
// B staged K-pair-interleaved: bw[(k>>1)*(2*BN) + 2*n + (k&1)] = W[k][n].
// ---------------------------------------------------------------------------
template <int KSTEPS, int BN, int NT>
__device__ __forceinline__ void mm_row(const float* abuf, int astride,
                                       const float* bw, int ml, int hi,
                                       v8f acc[NT]) {
  const v8f zero = {0.f, 0.f, 0.f, 0.f, 0.f, 0.f, 0.f, 0.f};
#pragma unroll
  for (int t = 0; t < NT; ++t) acc[t] = zero;
#pragma unroll 2
  for (int k = 0; k < KSTEPS; ++k) {
    v2f a = *(const v2f*)(abuf + ml * astride + 4 * k + 2 * hi);
    const float* brow = bw + (2 * k + hi) * (2 * BN) + 2 * ml;
#pragma unroll
    for (int t = 0; t < NT; ++t) {
      v2f b = *(const v2f*)(brow + 32 * t);
      acc[t] = __builtin_amdgcn_wmma_f32_16x16x4_f32(
          false, a, false, b, (short)0, acc[t], false, false);
    }
  }
}

// global row-major W[K][N] -> LDS K-pair-interleaved layout
__device__ __forceinline__ void fill_pairs(float* dst, const float* src,
                                           int K, int N, int tid, int nthr) {
  for (int idx = tid; idx < K * N; idx += nthr) {
    int k = idx / N;
    int n = idx - k * N;
    dst[(k >> 1) * (2 * N) + 2 * n + (k & 1)] = src[idx];
  }
}

__global__ void zero_kernel(float* __restrict__ p, size_t n) {
  size_t i = (size_t)blockIdx.x * blockDim.x + threadIdx.x;
  size_t s = (size_t)gridDim.x * blockDim.x;
  for (; i < n; i += s) p[i] = 0.0f;
}

extern "C" __global__ void __launch_bounds__(TPB, 1)
lemmoe_main(const int* __restrict__ edge_index,     // (2,E), row 0 = dst node
            const float* __restrict__ edge_sh,      // (E,9)
            const float* __restrict__ edge_length,  // (E)
            const float* __restrict__ one_hot,      // (E,128)
            const float* __restrict__ W1,           // (136,128)
            const float* __restrict__ W2,           // (128,128)
            const float* __restrict__ W3,           // (128,128)
            const float* __restrict__ Wenv,         // (128,96)
            float* __restrict__ d_lat,              // (E,128)
            float* __restrict__ d_nf,               // (N,288)  pre-zeroed
            float* __restrict__ d_ef,               // (E,288)
            float* __restrict__ d_cut)              // (E)
{
  extern __shared__ float smem[];
  // weights: 68*256 + 64*256 + 64*256 + 64*192 = 62464 floats (244 KB)
  float* w1p = smem;
  float* w2p = w1p + 68 * 256;
  float* w3p = w2p + 64 * 256;
  float* wep = w3p + 64 * 256;
  const int tid  = threadIdx.x;
  const int wave = tid >> 5;
  const int lane = tid & 31;
  const int ml   = lane & 15;
  const int hi   = lane >> 4;
  // per-wave staging: 16x136 + 16x128 + 16 + 144 = 4384 floats
  float* wbase = wep + 64 * 192 + wave * 4384;
  float* bufX  = wbase;            // 16 x DIN (also reused as 16x96 weights)
  float* bufH  = bufX + 16 * DIN;  // 16 x 128
  float* auxc  = bufH + 16 * 128;  // 16 cutoff scales
  float* auxsh = auxc + 16;        // 16 x 9 spherical harmonics

  fill_pairs(w1p, W1, 136, 128, tid, TPB);
  fill_pairs(w2p, W2, 128, 128, tid, TPB);
  fill_pairs(w3p, W3, 128, 128, tid, TPB);
  fill_pairs(wep, Wenv, 128, 96, tid, TPB);
  __syncthreads();

  const int gw = blockIdx.x * WAVES_PB + wave;
  const int nw = gridDim.x * WAVES_PB;
  const float pref = 0.632455532f;           // sqrt(2/R_MAX)
  const float PI_F = 3.14159265358979f;
  const float nscale = 0.2f;                 // AVG_NEIGH^-0.5

#if HAVE_TDM
  const unsigned ldsX  = (unsigned)(uintptr_t)bufX;
  const unsigned ldsSH = (unsigned)(uintptr_t)auxsh;
  // one_hot tile: 16 rows x 128 f32, LDS row stride 136 f32 -> pad 8 DWORDs
  // every 128 DWORDs: pad_interval=6 (128 DW), pad_amount=7 (8 DW)
  const unsigned flagsOH = (2u << 16) | (1u << 20) | (6u << 22) | (7u << 25);
  const unsigned flagsSH = (2u << 16);  // contiguous 144 f32, no padding
#endif

  for (int tile = gw; tile < NTILES; tile += nw) {
    const int e16 = tile * 16;

#if HAVE_TDM
    // make sure previous tile's LDS reads of bufX/auxsh have completed
    asm volatile("s_wait_dscnt 0x0" ::: "memory");
    tdm_load_2d(ldsX, one_hot + (size_t)e16 * ONEHOT, flagsOH,
                /*tdim0=*/ONEHOT, /*tdim1=*/16,
                /*tile0=*/ONEHOT, /*tile1=*/16, /*stride0=*/ONEHOT);
    tdm_load_2d(ldsSH, edge_sh + (size_t)e16 * 9, flagsSH,
                /*tdim0=*/144, /*tdim1=*/1,
                /*tile0=*/144, /*tile1=*/1, /*stride0=*/144);
#endif

    if (tile + nw < NTILES) {  // warm L2 for the next tile's one-hot rows
      const int ne = (tile + nw) * 16 + ml;
      __builtin_prefetch(one_hot + (size_t)ne * ONEHOT, 0, 1);
    }

    // ---- per-edge scalars: bessel invariants + polynomial cutoff ----
    // (writes bufX cols 128..135 = the TDM pad region: disjoint from DMA)
    {
      const int e = e16 + ml;
      float r    = edge_length[e];
      float xr   = r * (1.0f / 5.0f);        // r / R_MAX
      float invr = fast_rcp(r);
      float x2 = xr * xr, x3 = x2 * xr, x6 = x3 * x3;
      float cut = 1.0f - 28.0f * x6 + 48.0f * x6 * xr - 21.0f * x6 * x2;
      cut = (xr < 1.0f) ? cut : 0.0f;
      float cs = (cut > 0.0f) ? cut : 0.0f;
      if (lane < 16) {
        auxc[ml] = cs;
        __builtin_nontemporal_store(cut, d_cut + e);
#pragma unroll
        for (int j = 0; j < NBASIS; ++j) {
          float f = (float)(j + 1) * PI_F;
          bufX[ml * DIN + ONEHOT + j] = pref * __sinf(f * xr) * invr;
        }
      }
    }

#if HAVE_TDM
    __builtin_amdgcn_s_wait_tensorcnt(0);
    asm volatile("" ::: "memory");
#else
    for (int m = 0; m < 16; ++m) {
      v4f v = ((const v4f*)(one_hot + (size_t)(e16 + m) * ONEHOT))[lane];
      *(v4f*)(bufX + m * DIN + lane * 4) = v;
    }
    for (int idx = lane; idx < 144; idx += 32) {
      int m = idx / 9, j = idx - m * 9;
      auxsh[idx] = edge_sh[(size_t)(e16 + m) * 9 + j];
    }
#endif

    // ---- layer 1: H1 = silu(X @ W1), K=136 ----
    {
      v8f acc[8];
      mm_row<34, 128, 8>(bufX, DIN, w1p, ml, hi, acc);
#pragma unroll
      for (int t = 0; t < 8; ++t)
#pragma unroll
        for (int r = 0; r < 8; ++r) {
          float v = acc[t][r];
          v = v * fast_rcp(1.0f + __expf(-v));   // silu, hw v_rcp_f32
          bufH[(r + 8 * hi) * 128 + t * 16 + ml] = v;
        }
    }

    // ---- layer 2: H2 = silu(H1 @ W2), K=128 ----
    {
      v8f acc[8];
      mm_row<32, 128, 8>(bufH, 128, w2p, ml, hi, acc);
#pragma unroll
      for (int t = 0; t < 8; ++t)
#pragma unroll
        for (int r = 0; r < 8; ++r) {
          float v = acc[t][r];
          v = v * fast_rcp(1.0f + __expf(-v));   // silu, hw v_rcp_f32
          bufX[(r + 8 * hi) * DIN + t * 16 + ml] = v;
        }
    }

    // ---- layer 3: latents = cutoff * (H2 @ W3) -> bufH ----
    {
      v8f acc[8];
      mm_row<32, 128, 8>(bufX, DIN, w3p, ml, hi, acc);
#pragma unroll
      for (int t = 0; t < 8; ++t)
#pragma unroll
        for (int r = 0; r < 8; ++r) {
          int row = r + 8 * hi;
          bufH[row * 128 + t * 16 + ml] = acc[t][r] * auxc[row];
        }
    }

    // ---- stream latents tile to global as 128-bit nontemporal stores ----
#pragma unroll
    for (int q = 0; q < 16; ++q) {
      int idx  = q * 32 + lane;        // 512 v4f slots in 16x128 tile
      int row  = idx >> 3;
      int col4 = idx & 7;
      v4f v = *(const v4f*)(bufH + row * 128 + col4 * 4);
      __builtin_nontemporal_store(
          v, (v4f*)(d_lat + (size_t)(e16 + row) * LATENT) + col4);
    }

    // ---- layer 4: weights = latents @ Wenv (16 x 96) -> bufX ----
    {
      v8f acc[6];
      mm_row<32, 96, 6>(bufH, 128, wep, ml, hi, acc);
#pragma unroll
      for (int t = 0; t < 6; ++t)
#pragma unroll
        for (int r = 0; r < 8; ++r)
          bufX[(r + 8 * hi) * WENV_N + t * 16 + ml] = acc[t][r];
    }

    // ---- edge features (outer product w x sh) + node scatter ----
    for (int m = 0; m < 16; ++m) {
      const int e = e16 + m;
      const int node = edge_index[e];          // edge_index[0][e]
      const float* wm  = bufX + m * WENV_N;
      const float* shm = auxsh + m * 9;
#pragma unroll
      for (int i = 0; i < 9; ++i) {
        int c = lane + 32 * i;                 // 0..287
        int wi, si;
        if (c < 32)       { wi = c;                si = 0; }
        else if (c < 128) { int t = c - 32;  int q = t / 3; wi = 32 + q; si = 1 + t - 3 * q; }
        else              { int t = c - 128; int q = t / 5; wi = 64 + q; si = 4 + t - 5 * q; }
        float v = wm[wi] * shm[si];
        __builtin_nontemporal_store(v, d_ef + (size_t)e * EFDIM + c);
        unsafeAtomicAdd(d_nf + (size_t)node * EFDIM + c, v * nscale);
      }
    }
  }
}

extern "C" void kernel_launch(void* const* d_in, const int* in_sizes, int n_in,
                              void* d_out, int out_size, void* d_ws, size_t ws_size,
                              hipStream_t stream) {
  // setup_inputs order:
  // 0 edge_index (2,E) i32 | 1 atom_type | 2 bond_type | 3 edge_sh (E,9) f32
  // 4 edge_length (E) f32  | 5 edge_one_hot (E,128) f32 | 6 bessel_freqs
  // 7 W1 (136,128) | 8 W2 (128,128) | 9 W3 (128,128) | 10 W_env (128,96)
  const int*   edge_index = (const int*)  d_in[0];
  const float* edge_sh    = (const float*)d_in[3];
  const float* edge_len   = (const float*)d_in[4];
  const float* one_hot    = (const float*)d_in[5];
  const float* W1 = (const float*)d_in[7];
  const float* W2 = (const float*)d_in[8];
  const float* W3 = (const float*)d_in[9];
  const float* We = (const float*)d_in[10];

  float* out   = (float*)d_out;
  float* d_lat = out;                                  // E*128
  float* d_nf  = d_lat + (size_t)N_EDGES * LATENT;     // N*288
  float* d_ef  = d_nf + (size_t)N_NODES * EFDIM;       // E*288
  float* d_cut = d_ef + (size_t)N_EDGES * EFDIM;       // E

  zero_kernel<<<1024, 256, 0, stream>>>(d_nf, (size_t)N_NODES * EFDIM);

  // LDS: 62464 (weights) + 4*4384 (wave staging) = 80000 floats = 320000 B
  const size_t smem_bytes = 320000;
  lemmoe_main<<<NBLOCKS, TPB, smem_bytes, stream>>>(
      edge_index, edge_sh, edge_len, one_hot, W1, W2, W3, We,
      d_lat, d_nf, d_ef, d_cut);
}